// GIN_20890720928313
// MI455X (gfx1250) — compile-verified
//
#include <hip/hip_runtime.h>

typedef __attribute__((ext_vector_type(2))) float v2f;
typedef __attribute__((ext_vector_type(8))) float v8f;

#define GIN_N 50000
#define GIN_E 1600000
#define GIN_H 64
#define GIN_L 3
#define GIN_G 128
#define BN_EPS 1e-5f

// tiles: 3125 (M) x 4 (N) = 12500 waves per GEMM
#define M_TILES 3125
#define TILES_TOTAL (M_TILES * 4)
#define WAVES_PER_BLOCK 8
#define GEMM_BLOCKS ((TILES_TOTAL + WAVES_PER_BLOCK - 1) / WAVES_PER_BLOCK)

// ---------------------------------------------------------------------------
// Edge scatter-add: agg[dst] += h[src]   (agg pre-initialized with h itself,
// which realizes (1+eps)*x + sum_neighbors with eps=0)
// One float4 per thread, 16 threads per edge -> coalesced 256B gather per edge.
// ---------------------------------------------------------------------------
__global__ __launch_bounds__(256) void gin_edge_scatter(
    const float* __restrict__ h, const int* __restrict__ src,
    const int* __restrict__ dst, float* __restrict__ agg) {
  long gid = (long)blockIdx.x * blockDim.x + threadIdx.x;
  if (gid >= (long)GIN_E * 16) return;
  int e  = (int)(gid >> 4);
  int c4 = (int)(gid & 15) << 2;
  int s = src[e];
  int d = dst[e];
  const float4 v = *reinterpret_cast<const float4*>(h + (long)s * GIN_H + c4);
  float* out = agg + (long)d * GIN_H + c4;
  atomicAdd(out + 0, v.x);
  atomicAdd(out + 1, v.y);
  atomicAdd(out + 2, v.z);
  atomicAdd(out + 3, v.w);
}

// ---------------------------------------------------------------------------
// GEMM1: Z = A @ W + b  using V_WMMA_F32_16X16X4_F32, fused BN statistics
// (column sums / sum-of-squares reduced via LDS atomics, then global atomics).
// A: [N,64] row-major, W: [64,64] row-major (k,n), Z: [N,64].
// ---------------------------------------------------------------------------
__global__ __launch_bounds__(256) void gin_gemm1_stats(
    const float* __restrict__ A, const float* __restrict__ W,
    const float* __restrict__ bias, float* __restrict__ Z,
    float* __restrict__ colsum, float* __restrict__ colsumsq) {
  __shared__ float s_sum[GIN_H];
  __shared__ float s_sq[GIN_H];
  int tid = threadIdx.x;
  if (tid < GIN_H) { s_sum[tid] = 0.0f; s_sq[tid] = 0.0f; }
  __syncthreads();

  int wave = tid >> 5;
  int lane = tid & 31;
  int tile = blockIdx.x * WAVES_PER_BLOCK + wave;
  bool active = tile < TILES_TOTAL;   // wave-uniform -> EXEC all-ones inside

  if (active) {
    int tm = tile >> 2;
    int tn = tile & 3;
    int rowBase = tm * 16;
    int colBase = tn * 16;
    int lh = lane & 15;      // row (A) / col (B,C) within tile
    int hi = lane >> 4;      // K-half selector

    v8f acc = {};
    const float* arow = A + (long)(rowBase + lh) * GIN_H + 2 * hi;
#pragma unroll
    for (int k0 = 0; k0 < GIN_H; k0 += 4) {
      v2f a, b;
      a.x = arow[k0];
      a.y = arow[k0 + 1];
      b.x = W[(k0 + 2 * hi) * GIN_H + colBase + lh];
      b.y = W[(k0 + 2 * hi + 1) * GIN_H + colBase + lh];
      acc = __builtin_amdgcn_wmma_f32_16x16x4_f32(
          false, a, false, b, (short)0, acc, false, false);
    }

    float bcol = bias[colBase + lh];
    float s = 0.0f, q = 0.0f;
    float* zt = Z + (long)(rowBase + hi * 8) * GIN_H + colBase + lh;
#pragma unroll
    for (int r = 0; r < 8; ++r) {
      float v = acc[r] + bcol;
      zt[(long)r * GIN_H] = v;
      s += v;
      q += v * v;
    }
    atomicAdd(&s_sum[colBase + lh], s);   // ds_add_f32
    atomicAdd(&s_sq[colBase + lh], q);
  }
  __syncthreads();
  if (tid < GIN_H) {
    atomicAdd(&colsum[tid], s_sum[tid]);
    atomicAdd(&colsumsq[tid], s_sq[tid]);
  }
}

// ---------------------------------------------------------------------------
// Finalize BN: scale = rsqrt(var+eps)*gamma, shift = beta - mu*scale
// ---------------------------------------------------------------------------
__global__ void gin_bn_finalize(const float* __restrict__ colsum,
                                const float* __restrict__ colsumsq,
                                const float* __restrict__ gamma,
                                const float* __restrict__ beta,
                                float* __restrict__ scale,
                                float* __restrict__ shift) {
  int t = threadIdx.x;
  if (t >= GIN_H) return;
  float inv_n = 1.0f / (float)GIN_N;
  float mu  = colsum[t] * inv_n;
  float var = colsumsq[t] * inv_n - mu * mu;
  float rstd = rsqrtf(var + BN_EPS);
  float sc = rstd * gamma[t];
  scale[t] = sc;
  shift[t] = beta[t] - mu * sc;
}

// ---------------------------------------------------------------------------
// GEMM2: Hout = relu( relu(BN(Z)) @ W2 + b2 ).  BN+ReLU folded into the
// A-operand load: a = max(z*scale[k] + shift[k], 0).
// ---------------------------------------------------------------------------
__global__ __launch_bounds__(256) void gin_gemm2_bn(
    const float* __restrict__ Z, const float* __restrict__ scale,
    const float* __restrict__ shift, const float* __restrict__ W,
    const float* __restrict__ bias, float* __restrict__ Hout) {
  int tid = threadIdx.x;
  int wave = tid >> 5;
  int lane = tid & 31;
  int tile = blockIdx.x * WAVES_PER_BLOCK + wave;
  if (tile >= TILES_TOTAL) return;   // wave-uniform

  int tm = tile >> 2;
  int tn = tile & 3;
  int rowBase = tm * 16;
  int colBase = tn * 16;
  int lh = lane & 15;
  int hi = lane >> 4;

  v8f acc = {};
  const float* zrow = Z + (long)(rowBase + lh) * GIN_H + 2 * hi;
#pragma unroll
  for (int k0 = 0; k0 < GIN_H; k0 += 4) {
    int k = k0 + 2 * hi;
    v2f a, b;
    a.x = fmaxf(zrow[k0]     * scale[k]     + shift[k],     0.0f);
    a.y = fmaxf(zrow[k0 + 1] * scale[k + 1] + shift[k + 1], 0.0f);
    b.x = W[k * GIN_H + colBase + lh];
    b.y = W[(k + 1) * GIN_H + colBase + lh];
    acc = __builtin_amdgcn_wmma_f32_16x16x4_f32(
        false, a, false, b, (short)0, acc, false, false);
  }

  float bcol = bias[colBase + lh];
  float* ht = Hout + (long)(rowBase + hi * 8) * GIN_H + colBase + lh;
#pragma unroll
  for (int r = 0; r < 8; ++r) {
    ht[(long)r * GIN_H] = fmaxf(acc[r] + bcol, 0.0f);   // outer GIN relu
  }
}

// ---------------------------------------------------------------------------
// Global mean pool (sum + counts via atomics)
// ---------------------------------------------------------------------------
__global__ __launch_bounds__(256) void gin_pool(
    const float* __restrict__ h, const int* __restrict__ batch,
    float* __restrict__ pooled, float* __restrict__ cnt) {
  long gid = (long)blockIdx.x * blockDim.x + threadIdx.x;
  if (gid >= (long)GIN_N * GIN_H) return;
  int n = (int)(gid >> 6);
  int c = (int)(gid & 63);
  int g = batch[n];
  atomicAdd(&pooled[(long)g * GIN_H + c], h[gid]);
  if (c == 0) atomicAdd(&cnt[g], 1.0f);
}

// ---------------------------------------------------------------------------
// Head MLP part 1: t = relu(mean_pooled @ Wh1 + bh1)   [128 x 64]
// ---------------------------------------------------------------------------
__global__ __launch_bounds__(256) void gin_head1(
    const float* __restrict__ pooled, const float* __restrict__ cnt,
    const float* __restrict__ Wh1, const float* __restrict__ bh1,
    float* __restrict__ t) {
  int gid = blockIdx.x * blockDim.x + threadIdx.x;
  if (gid >= GIN_G * GIN_H) return;
  int g = gid >> 6;
  int c = gid & 63;
  float inv = 1.0f / fmaxf(cnt[g], 1.0f);
  float s = bh1[c];
#pragma unroll 8
  for (int k = 0; k < GIN_H; ++k) {
    s += pooled[(long)g * GIN_H + k] * inv * Wh1[k * GIN_H + c];
  }
  t[gid] = fmaxf(s, 0.0f);
}

// Head MLP part 2: out[g] = t[g,:] @ Wh2 + bh2   [128 x 1]
__global__ void gin_head2(const float* __restrict__ t,
                          const float* __restrict__ Wh2,
                          const float* __restrict__ bh2,
                          float* __restrict__ out) {
  int g = threadIdx.x;
  if (g >= GIN_G) return;
  float s = bh2[0];
#pragma unroll 8
  for (int c = 0; c < GIN_H; ++c) s += t[g * GIN_H + c] * Wh2[c];
  out[g] = s;
}

// ---------------------------------------------------------------------------
// Host-side orchestration
// ---------------------------------------------------------------------------
extern "C" void kernel_launch(void* const* d_in, const int* in_sizes, int n_in,
                              void* d_out, int out_size, void* d_ws, size_t ws_size,
                              hipStream_t stream) {
  const float* x     = (const float*)d_in[0];
  const int*   ei    = (const int*)d_in[1];   // [2, E] flattened
  const int*   batch = (const int*)d_in[2];
  const float* W1    = (const float*)d_in[3]; // [L,64,64]
  const float* b1    = (const float*)d_in[4]; // [L,64]
  const float* gam   = (const float*)d_in[5];
  const float* bet   = (const float*)d_in[6];
  const float* W2    = (const float*)d_in[7];
  const float* b2    = (const float*)d_in[8];
  const float* Wh1   = (const float*)d_in[9];
  const float* bh1   = (const float*)d_in[10];
  const float* Wh2   = (const float*)d_in[11];
  const float* bh2   = (const float*)d_in[12];

  const int* src = ei;
  const int* dst = ei + GIN_E;

  const size_t NODE_BYTES = (size_t)GIN_N * GIN_H * sizeof(float); // 12.8 MB
  char* ws = (char*)d_ws;
  float* agg      = (float*)(ws);
  float* z        = (float*)(ws + NODE_BYTES);
  float* hb0      = (float*)(ws + 2 * NODE_BYTES);
  float* hb1      = (float*)(ws + 3 * NODE_BYTES);
  char*  small    = ws + 4 * NODE_BYTES;
  float* colsum   = (float*)(small);            // 64 f
  float* colsumsq = (float*)(small + 256);      // 64 f
  float* scale    = (float*)(small + 512);      // 64 f
  float* shiftv   = (float*)(small + 768);      // 64 f
  float* pooled   = (float*)(small + 1024);     // 128*64 f = 32768 B
  float* cnt      = (float*)(small + 1024 + 32768);         // 128 f
  float* thead    = (float*)(small + 1024 + 32768 + 512);   // 128*64 f

  const int edgeThreads = GIN_E * 16;
  const int edgeBlocks  = (edgeThreads + 255) / 256;
  const int poolBlocks  = ((GIN_N * GIN_H) + 255) / 256;

  const float* hin = x;
  for (int l = 0; l < GIN_L; ++l) {
    float* hout = (l == 0) ? hb0 : ((l == 1) ? hb1 : hb0);
    const float* Wl1 = W1 + (size_t)l * GIN_H * GIN_H;
    const float* Wl2 = W2 + (size_t)l * GIN_H * GIN_H;

    // agg = h  (self term), then scatter-add neighbors
    hipMemcpyAsync(agg, hin, NODE_BYTES, hipMemcpyDeviceToDevice, stream);
    gin_edge_scatter<<<edgeBlocks, 256, 0, stream>>>(hin, src, dst, agg);

    hipMemsetAsync(colsum, 0, 512, stream);  // colsum + colsumsq
    gin_gemm1_stats<<<GEMM_BLOCKS, 256, 0, stream>>>(
        agg, Wl1, b1 + l * GIN_H, z, colsum, colsumsq);
    gin_bn_finalize<<<1, 64, 0, stream>>>(
        colsum, colsumsq, gam + l * GIN_H, bet + l * GIN_H, scale, shiftv);
    gin_gemm2_bn<<<GEMM_BLOCKS, 256, 0, stream>>>(
        z, scale, shiftv, Wl2, b2 + l * GIN_H, hout);
    hin = hout;
  }

  hipMemsetAsync(pooled, 0, 32768 + 512, stream);  // pooled + cnt (adjacent)
  gin_pool<<<poolBlocks, 256, 0, stream>>>(hin, batch, pooled, cnt);
  gin_head1<<<(GIN_G * GIN_H + 255) / 256, 256, 0, stream>>>(
      pooled, cnt, Wh1, bh1, thead);
  gin_head2<<<1, GIN_G, 0, stream>>>(thead, Wh2, bh2, (float*)d_out);
}